// ICSModel_45758581571859
// MI455X (gfx1250) — compile-verified
//
#include <hip/hip_runtime.h>
#include <stdint.h>

#define NF    128     // features per row
#define NG    32      // categorical groups (one per wave32 lane)
#define EMB   8
#define OUTW  352     // 32*11 output columns
#define TABF  (NG * 4 * EMB)   // 1024 floats = 4 KB of embedding tables
#define WAVES_PER_BLOCK 8

typedef uint32_t u32x4 __attribute__((ext_vector_type(4)));
typedef int32_t  i32x4 __attribute__((ext_vector_type(4)));
typedef int32_t  i32x8 __attribute__((ext_vector_type(8)));

__global__ __launch_bounds__(256)
void ics_embed_kernel(const int*   __restrict__ unscaled,
                      const float* __restrict__ scaled,
                      const float* __restrict__ tables,
                      float*       __restrict__ out,
                      int rows)
{
    __shared__ __align__(16) float lds_tab[TABF];
    __shared__ __align__(16) float lds_row[WAVES_PER_BLOCK * OUTW];

    const int tid  = threadIdx.x;
    const int lane = tid & 31;
    const int wave = tid >> 5;

    // ---- Stage the 4 KB embedding tables into LDS ----------------------
#if defined(__gfx1250__) && __has_builtin(__builtin_amdgcn_tensor_load_to_lds)
    if (wave == 0) {
        // Tensor DMA descriptor (D#): 1D copy of 1024 fp32, expressed as a
        // 1024x1 tile of a 1024x1 tensor, 4-byte elements.
        const uint64_t ga       = (uint64_t)(uintptr_t)tables;
        const uint32_t lds_addr = (uint32_t)(uintptr_t)&lds_tab[0]; // low 32 bits == LDS byte offset

        u32x4 g0;
        g0[0] = 1u;                                   // count=1, user mode, no gather
        g0[1] = lds_addr;                             // lds_addr [63:32]
        g0[2] = (uint32_t)ga;                         // global_addr lo
        g0[3] = (uint32_t)((ga >> 32) & 0x01FFFFFFu)  // global_addr [56:32]
              | (2u << 30);                           // type = 2 ("image")

        i32x8 g1;
        g1[0] = (int)(2u << 16);        // workgroup_mask=0 (not in cluster), data_size=2 (4B)
        g1[1] = (int)((uint32_t)TABF << 16); // atomic_barrier_addr=0 | tensor_dim0[15:0]=1024
        g1[2] = (int)(1u << 16);        // tensor_dim0[31:16]=0 | tensor_dim1[15:0]=1
        g1[3] = (int)((uint32_t)TABF << 16); // tensor_dim1[31:16]=0 | tile_dim0=1024
        g1[4] = 1;                      // tile_dim1=1, tile_dim2=0 (unused)
        g1[5] = TABF;                   // tensor_dim0_stride lo = 1024
        g1[6] = 0;                      // stride0 hi | stride1 lo
        g1[7] = 0;                      // stride1 hi

        const i32x4 zero4 = {0, 0, 0, 0};
        const i32x8 zero8 = {0, 0, 0, 0, 0, 0, 0, 0};
        // 6-arg form (clang-23 / therock-10.0 headers): groups 2/3 unused
        // (<=2D tensor), trailing group + cpol zero.
        __builtin_amdgcn_tensor_load_to_lds(g0, g1, zero4, zero4, zero8, 0);
        __builtin_amdgcn_s_wait_tensorcnt(0);
    }
#else
    for (int i = tid; i < TABF; i += 256) lds_tab[i] = tables[i];
#endif
    __syncthreads();

    const int wavesTotal = gridDim.x * WAVES_PER_BLOCK;
    const int wid        = blockIdx.x * WAVES_PER_BLOCK + wave;
    float* myrow = &lds_row[wave * OUTW];

    for (int row = wid; row < rows; row += wavesTotal) {
        const size_t fbase = (size_t)row * NF;

        // Lane g handles group g: feature 4g is categorical, 4g+1..3 continuous.
        const int    id = unscaled[fbase + (size_t)(lane * 4)];
        const float4 s4 = *(const float4*)(scaled + fbase + (size_t)(lane * 4));

        // Prefetch next iteration's inputs (global_prefetch_b8).
        if (row + wavesTotal < rows) {
            const size_t nbase = fbase + (size_t)wavesTotal * NF;
            __builtin_prefetch(scaled   + nbase + (size_t)(lane * 4), 0, 0);
            __builtin_prefetch(unscaled + nbase + (size_t)(lane * 4), 0, 0);
        }

        // Embedding lookup from LDS: table[g][id][0..7], 16B-aligned b128 loads.
        const float4* e  = (const float4*)&lds_tab[lane * 32 + id * 8];
        const float4  e0 = e[0];
        const float4  e1 = e[1];

        // Stage this wave's output row in LDS (bank-conflict-free: 11 odd).
        float* dst = myrow + lane * 11;
        dst[0] = e0.x;  dst[1] = e0.y;  dst[2] = e0.z;  dst[3] = e0.w;
        dst[4] = e1.x;  dst[5] = e1.y;  dst[6] = e1.z;  dst[7] = e1.w;
        dst[8] = s4.y;  dst[9] = s4.z;  dst[10] = s4.w;

        // Intra-wave LDS exchange: DS ops are in-order per wave; just stop
        // the compiler from reordering across the exchange.
        __builtin_amdgcn_fence(__ATOMIC_RELEASE, "wavefront");
        __builtin_amdgcn_wave_barrier();
        __builtin_amdgcn_fence(__ATOMIC_ACQUIRE, "wavefront");

        // Fully coalesced writeback: 11 x (128B-per-wave) b32 stores.
        float* orow = out + (size_t)row * OUTW;
        #pragma unroll
        for (int k = 0; k < 11; ++k)
            orow[lane + 32 * k] = myrow[lane + 32 * k];

        // Keep next iteration's LDS stores from passing this iteration's reads.
        __builtin_amdgcn_fence(__ATOMIC_RELEASE, "wavefront");
        __builtin_amdgcn_wave_barrier();
    }
}

extern "C" void kernel_launch(void* const* d_in, const int* in_sizes, int n_in,
                              void* d_out, int out_size, void* d_ws, size_t ws_size,
                              hipStream_t stream) {
    (void)n_in; (void)out_size; (void)d_ws; (void)ws_size;
    const int*   unscaled = (const int*)d_in[0];    // [B,T,128] int32
    const float* scaled   = (const float*)d_in[1];  // [B,T,128] fp32
    const float* tables   = (const float*)d_in[2];  // [32,4,8]  fp32
    float*       out      = (float*)d_out;          // [B,T,352] fp32

    const int rows = in_sizes[1] / NF;              // B*T = 131072

    int blocks = 2048;                              // 16384 waves -> ~8 rows/wave
    const int maxBlocks = (rows + WAVES_PER_BLOCK - 1) / WAVES_PER_BLOCK;
    if (blocks > maxBlocks) blocks = maxBlocks;
    if (blocks < 1) blocks = 1;

    ics_embed_kernel<<<blocks, 256, 0, stream>>>(unscaled, scaled, tables, out, rows);
}